// SOM_89687507075387
// MI455X (gfx1250) — compile-verified
//
#include <hip/hip_runtime.h>

typedef __attribute__((ext_vector_type(2))) float v2f;
typedef __attribute__((ext_vector_type(4))) float v4f;
typedef __attribute__((ext_vector_type(8))) float v8f;

#define BATCH 2048
#define LM    2000
#define NF    64
#define LTILES (LM / 16)   // 125 exact
#define BSTRIPS (BATCH / 16)

// ---------------------------------------------------------------------------
// Stage 0: landmark squared norms (trivial, L2-resident input)
// ---------------------------------------------------------------------------
__global__ void SOM_lm2_kernel(const float* __restrict__ lm,
                               float* __restrict__ lm2) {
    int l = blockIdx.x * blockDim.x + threadIdx.x;
    if (l >= LM) return;
    const v4f* row = (const v4f*)(lm + (long long)l * NF);
    float s = 0.f;
#pragma unroll
    for (int i = 0; i < NF / 4; ++i) {
        v4f v = row[i];
        s += v.x * v.x + v.y * v.y + v.z * v.z + v.w * v.w;
    }
    lm2[l] = s;
}

// ---------------------------------------------------------------------------
// Stage 1: BMU (argmin) via V_WMMA_F32_16X16X4_F32.
// Each wave owns one 16-row strip of x; A fragments (16 K-steps x 2 VGPRs)
// are loaded once and reused across all 125 landmark tiles.
// score = lm2[l] - 2*(x . lm[l])   (x^2 row term dropped: constant per row)
// ---------------------------------------------------------------------------
__global__ void __launch_bounds__(256)
SOM_bmu_kernel(const float* __restrict__ x, const float* __restrict__ lm,
               const float* __restrict__ lm2, int* __restrict__ bmu) {
    const int lane  = threadIdx.x & 31;
    const int wave  = threadIdx.x >> 5;
    const int strip = blockIdx.x * 8 + wave;   // 0..127
    const int r0    = strip * 16;
    const int half  = lane >> 4;               // 0: K=0,1 / rows 0-7 ; 1: K=2,3 / rows 8-15
    const int mn    = lane & 15;               // A: row index ; B/C: column index

    // A-matrix fragments: lane holds x[r0+mn][4*kk + 2*half + {0,1}]
    v2f afrag[16];
    const float* xrow = x + (long long)(r0 + mn) * NF + half * 2;
#pragma unroll
    for (int kk = 0; kk < 16; ++kk)
        afrag[kk] = *(const v2f*)(xrow + kk * 4);

    float bestv[8];
    int   besti[8];
#pragma unroll
    for (int r = 0; r < 8; ++r) { bestv[r] = 3.402823e38f; besti[r] = 0; }

    for (int t = 0; t < LTILES; ++t) {
        const int lcol = t * 16 + mn;          // landmark column for this lane
        const float* lrow = lm + (long long)lcol * NF + half * 2;
        v8f acc = {};
#pragma unroll
        for (int kk = 0; kk < 16; ++kk) {
            v2f bfrag = *(const v2f*)(lrow + kk * 4);   // B[k][n] = lm[n][k]
            acc = __builtin_amdgcn_wmma_f32_16x16x4_f32(
                false, afrag[kk], false, bfrag, (short)0, acc, false, false);
        }
        const float l2 = lm2[lcol];
#pragma unroll
        for (int r = 0; r < 8; ++r) {
            float s = l2 - 2.0f * acc[r];
            if (s < bestv[r]) { bestv[r] = s; besti[r] = lcol; }
        }
    }

    // min+index reduction across the 16 lanes sharing each row
    // (xor masks 1,2,4,8 never cross the lane16 boundary)
#pragma unroll
    for (int m = 1; m <= 8; m <<= 1) {
#pragma unroll
        for (int r = 0; r < 8; ++r) {
            float ov = __shfl_xor(bestv[r], m, 32);
            int   oi = __shfl_xor(besti[r], m, 32);
            if (ov < bestv[r] || (ov == bestv[r] && oi < besti[r])) {
                bestv[r] = ov; besti[r] = oi;
            }
        }
    }
    if (mn == 0) {
#pragma unroll
        for (int r = 0; r < 8; ++r)
            bmu[r0 + half * 8 + r] = besti[r];   // lane0: rows 0-7, lane16: rows 8-15
    }
}

// ---------------------------------------------------------------------------
// Stage 2: delta0[b,l,f] = qd[bmu[b],l] * (x[b,f] - lm[l,f])
// 1.05 GB of streaming output -> bandwidth bound. Non-temporal b128 stores
// keep lm/qd resident in L2 while the output streams through.
// ---------------------------------------------------------------------------
__global__ void __launch_bounds__(256)
SOM_delta_kernel(const float* __restrict__ x, const float* __restrict__ lm,
                 const float* __restrict__ qd, const int* __restrict__ bmu,
                 float* __restrict__ out) {
    const int b  = blockIdx.x;          // 0..2047
    const int t  = blockIdx.y;          // 0..124
    const int li = threadIdx.x >> 4;    // landmark within tile, 0..15
    const int fi = threadIdx.x & 15;    // float4 feature chunk, 0..15
    const int l  = t * 16 + li;

    const int   idx = bmu[b];                              // uniform -> scalar load
    const float h   = qd[(long long)idx * LM + l];

    v4f xv = *(const v4f*)(x + (long long)b * NF + fi * 4);
    v4f lv = *(const v4f*)(lm + (long long)l * NF + fi * 4);
    v4f o;
    o.x = h * (xv.x - lv.x);
    o.y = h * (xv.y - lv.y);
    o.z = h * (xv.z - lv.z);
    o.w = h * (xv.w - lv.w);

    float* dst = out + ((long long)b * LM + l) * NF + fi * 4;
    __builtin_nontemporal_store(o, (v4f*)dst);
}

// ---------------------------------------------------------------------------
extern "C" void kernel_launch(void* const* d_in, const int* in_sizes, int n_in,
                              void* d_out, int out_size, void* d_ws, size_t ws_size,
                              hipStream_t stream) {
    const float* x  = (const float*)d_in[0];   // (2048, 64)
    const float* lm = (const float*)d_in[1];   // (2000, 64)
    const float* qd = (const float*)d_in[2];   // (2000, 2000)
    float* out = (float*)d_out;                // (2048, 2000, 64)

    int*   bmu = (int*)d_ws;                          // 2048 ints
    float* lm2 = (float*)((char*)d_ws + 8192);        // 2000 floats

    SOM_lm2_kernel<<<(LM + 255) / 256, 256, 0, stream>>>(lm, lm2);
    SOM_bmu_kernel<<<BSTRIPS / 8, 256, 0, stream>>>(x, lm, lm2, bmu);

    dim3 grid(BATCH, LTILES);
    SOM_delta_kernel<<<grid, 256, 0, stream>>>(x, lm, qd, bmu, out);
}